// xDFM_31808527794326
// MI455X (gfx1250) — compile-verified
//
#include <hip/hip_runtime.h>

// xDeepFM forward for MI455X (gfx1250), fp32 end-to-end via V_WMMA_F32_16X16X4_F32.
// ~15 GFLOP / ~17 MB: latency-bound at 23.3 TB/s, so keep fp32 precision and
// focus on fusion, branch-free WMMA inner loops, and single-pass data movement.

typedef __attribute__((ext_vector_type(2))) float v2f;
typedef __attribute__((ext_vector_type(8))) float v8f;

#define B_SZ   2048
#define F_SZ   64
#define DRAW   32
#define D_EMB  15
#define ROWS   (B_SZ * F_SZ)       // 131072 (= B*F rows through emb/DNN)
#define U0     256
#define U1     128
#define DNNOUT 32
#define CONCAT (F_SZ + 50 + F_SZ * DNNOUT)   // 64 + 50 + 2048 = 2162

__device__ __forceinline__ v8f wmma4(v2f a, v2f b, v8f c) {
  // D = A(16x4,f32) * B(4x16,f32) + C(16x16,f32)
  return __builtin_amdgcn_wmma_f32_16x16x4_f32(
      /*neg_a=*/false, a, /*neg_b=*/false, b,
      /*c_mod=*/(short)0, c, /*reuse_a=*/false, /*reuse_b=*/false);
}

// --------------------------------------------------------------------------
// Kernel 1: fused embedding + linear term.
//   [ROWS x 32] @ [32 x 16]  where col 0..14 = emb_W, col 15 = liner_W.
//   Combined weight + bias staged in LDS once per block (uniform loads) so
//   the WMMA k-loop is branch-free (single base pointer for B).
//   One wave per 16-row tile; K=32 -> 8 WMMA k-steps.
//   Epilogue: unconditional embP stores (value-select pads col 15 with 0),
//   liner stores under a single ln==15 exec region (one mask toggle).
// --------------------------------------------------------------------------
__global__ void k_emb_liner(const float* __restrict__ x,
                            const float* __restrict__ embW,
                            const float* __restrict__ embB,
                            const float* __restrict__ linW,
                            const float* __restrict__ linB,
                            float* __restrict__ embP,
                            float* __restrict__ liner) {
  __shared__ float w16[DRAW * 16];   // combined [32][16]
  __shared__ float bias16[16];
  const int t = threadIdx.x;
  for (int i = t; i < DRAW * 16; i += 256) {
    const int k = i >> 4, n = i & 15;
    w16[i] = (n < D_EMB) ? embW[k * D_EMB + n] : linW[k];
  }
  if (t < 16) bias16[t] = (t < D_EMB) ? embB[t] : linB[0];

  const int lane = threadIdx.x & 31;
  const int wave = threadIdx.x >> 5;               // 8 waves / block
  const int row0 = (blockIdx.x * 8 + wave) * 16;
  const int ln   = lane & 15;                      // A row / B,C col
  const int sel  = lane >> 4;                      // k/m half selector

  // warm near caches for this wave's A rows while LDS fills
  __builtin_prefetch(&x[(size_t)(row0 + ln) * DRAW], 0, 3);
  __syncthreads();

  v8f c = {};
#pragma unroll
  for (int k0 = 0; k0 < DRAW; k0 += 4) {
    const int kk = k0 + 2 * sel;
    v2f a, b;
    a.x = x[(size_t)(row0 + ln) * DRAW + kk];
    a.y = x[(size_t)(row0 + ln) * DRAW + kk + 1];
    b.x = w16[kk * 16 + ln];
    b.y = w16[(kk + 1) * 16 + ln];
    c = wmma4(a, b, c);
  }

  const float bias    = bias16[ln];
  const bool  is_emb  = (ln < D_EMB);
  float vals[8];
#pragma unroll
  for (int r = 0; r < 8; ++r) vals[r] = c[r] + bias;

  // embP: all lanes store; col-15 lanes store the zero pad (v_cndmask, no branch)
#pragma unroll
  for (int r = 0; r < 8; ++r) {
    const int mr = r + 8 * sel;
    embP[(size_t)(row0 + mr) * 16 + ln] = is_emb ? vals[r] : 0.0f;
  }
  // liner: single exec region for the two ln==15 lanes
  if (!is_emb) {
#pragma unroll
    for (int r = 0; r < 8; ++r) liner[row0 + r + 8 * sel] = vals[r];
  }
}

// --------------------------------------------------------------------------
// Kernel 2: fused 3-layer DNN per 16-row tile.
//   16->256 (relu) -> 128 (relu) -> 32, all WMMA f32 16x16x4, LDS staging.
//   Block = 128 threads (4 waves), one tile per block.
// --------------------------------------------------------------------------
#define H1S 258   // 256 + 2: avoid 64-bank conflicts on A-operand reads
#define H2S 130

__global__ void k_dnn(const float* __restrict__ embP,
                      const float* __restrict__ W0, const float* __restrict__ b0,
                      const float* __restrict__ W1, const float* __restrict__ b1,
                      const float* __restrict__ Wo, const float* __restrict__ bo,
                      float* __restrict__ dnnout) {
  __shared__ float h1[16 * H1S];
  __shared__ float h2[16 * H2S];
  const int lane = threadIdx.x & 31;
  const int wave = threadIdx.x >> 5;   // 0..3
  const int row0 = blockIdx.x * 16;
  const int ln   = lane & 15;
  const int sel  = lane >> 4;

  // warm L2 for layer-1 weights while layer 0 runs (128 threads * 64 floats)
  __builtin_prefetch(W1 + (size_t)threadIdx.x * 64, 0, 1);

  // ---------------- layer 0: [16x16] x [16x256] ----------------
  {
    v8f acc[4] = {};
#pragma unroll
    for (int k0 = 0; k0 < 16; k0 += 4) {
      const int kk = k0 + 2 * sel;        // 0..14 (even)
      const int k1 = (kk + 1 < D_EMB) ? kk + 1 : D_EMB - 1;  // clamp: A col15==0
      v2f a;
      a.x = embP[(size_t)(row0 + ln) * 16 + kk];
      a.y = embP[(size_t)(row0 + ln) * 16 + kk + 1];
#pragma unroll
      for (int i = 0; i < 4; ++i) {
        const int n0 = (wave * 4 + i) * 16;
        v2f b;
        b.x = W0[kk * U0 + n0 + ln];
        b.y = W0[k1 * U0 + n0 + ln];
        acc[i] = wmma4(a, b, acc[i]);
      }
    }
#pragma unroll
    for (int i = 0; i < 4; ++i) {
      const int   n0   = (wave * 4 + i) * 16;
      const float bias = b0[n0 + ln];
#pragma unroll
      for (int r = 0; r < 8; ++r) {
        const int mr = r + 8 * sel;
        h1[mr * H1S + n0 + ln] = fmaxf(acc[i][r] + bias, 0.0f);
      }
    }
  }
  __syncthreads();

  // ---------------- layer 1: [16x256] x [256x128] ----------------
  {
    v8f acc[2] = {};
#pragma unroll 4
    for (int k0 = 0; k0 < U0; k0 += 4) {
      const int kk = k0 + 2 * sel;
      v2f a;
      a.x = h1[ln * H1S + kk];
      a.y = h1[ln * H1S + kk + 1];
#pragma unroll
      for (int i = 0; i < 2; ++i) {
        const int n0 = (wave * 2 + i) * 16;
        v2f b;
        b.x = W1[kk * U1 + n0 + ln];
        b.y = W1[(kk + 1) * U1 + n0 + ln];
        acc[i] = wmma4(a, b, acc[i]);
      }
    }
#pragma unroll
    for (int i = 0; i < 2; ++i) {
      const int   n0   = (wave * 2 + i) * 16;
      const float bias = b1[n0 + ln];
#pragma unroll
      for (int r = 0; r < 8; ++r) {
        const int mr = r + 8 * sel;
        h2[mr * H2S + n0 + ln] = fmaxf(acc[i][r] + bias, 0.0f);
      }
    }
  }
  __syncthreads();

  // ---------------- layer 2: [16x128] x [128x32] ----------------
  if (wave < 2) {                       // wave-uniform: EXEC stays all-ones
    v8f acc = {};
    const int n0 = wave * 16;
#pragma unroll 4
    for (int k0 = 0; k0 < U1; k0 += 4) {
      const int kk = k0 + 2 * sel;
      v2f a, b;
      a.x = h2[ln * H2S + kk];
      a.y = h2[ln * H2S + kk + 1];
      b.x = Wo[kk * DNNOUT + n0 + ln];
      b.y = Wo[(kk + 1) * DNNOUT + n0 + ln];
      acc = wmma4(a, b, acc);
    }
    const float bias = bo[n0 + ln];
#pragma unroll
    for (int r = 0; r < 8; ++r) {
      const int mr = r + 8 * sel;
      dnnout[(size_t)(row0 + mr) * DNNOUT + n0 + ln] = acc[r] + bias;
    }
  }
}

// --------------------------------------------------------------------------
// Kernel 3: CIN (5 layers), one block per batch sample, VALU.
//   Per-sample bilinear forms with N=10 / F_i=10 don't fill 16x16 tiles;
//   ~4 GFLOP total, fine on VALU. emb tile + h buffers in LDS; cin_W0
//   (163 KB) stays resident in the 192 MB L2 across all 2048 blocks.
// --------------------------------------------------------------------------
__global__ void k_cin(const float* __restrict__ embP,
                      const float* __restrict__ cw0, const float* __restrict__ cw1,
                      const float* __restrict__ cw2, const float* __restrict__ cw3,
                      const float* __restrict__ cw4,
                      float* __restrict__ cin) {
  __shared__ float e[F_SZ * 16];            // emb[b, f, d] (stride 16, col 15 unused)
  __shared__ float hbuf[2][10 * D_EMB];
  const int b = blockIdx.x;
  const int t = threadIdx.x;
  for (int i = t; i < F_SZ * 16; i += 256) e[i] = embP[(size_t)b * F_SZ * 16 + i];
  __syncthreads();

  const float* Ws[5] = {cw0, cw1, cw2, cw3, cw4};
  const int n = t / D_EMB;      // 0..9  (valid when t < 150)
  const int d = t - n * D_EMB;  // 0..14

  for (int li = 0; li < 5; ++li) {
    const float* W = Ws[li];
    if (t < 150) {
      float acc = 0.0f;
      if (li == 0) {
        for (int f = 0; f < F_SZ; ++f) {
          const float ef = e[f * 16 + d];
          const float* wr = W + (size_t)(f * F_SZ) * 10 + n;
          for (int g = 0; g < F_SZ; ++g)
            acc += ef * e[g * 16 + d] * wr[g * 10];
        }
      } else {
        const float* hc = hbuf[(li - 1) & 1];
        for (int f = 0; f < F_SZ; ++f) {
          const float ef = e[f * 16 + d];
          const float* wr = W + (size_t)(f * 10) * 10 + n;
          for (int g = 0; g < 10; ++g)
            acc += ef * hc[g * D_EMB + d] * wr[g * 10];
        }
      }
      hbuf[li & 1][n * D_EMB + d] = acc;
    }
    __syncthreads();
    if (t < 10) {                               // sum over embedding dim
      float s = 0.0f;
      for (int dd = 0; dd < D_EMB; ++dd) s += hbuf[li & 1][t * D_EMB + dd];
      cin[b * 50 + li * 10 + t] = s;
    }
    __syncthreads();
  }
}

// --------------------------------------------------------------------------
// Kernel 4: final dense: out[b] = feat(2162) . w + bias.
//   Deterministic LDS tree reduction (no float atomics).
// --------------------------------------------------------------------------
__global__ void k_final(const float* __restrict__ liner,
                        const float* __restrict__ cin,
                        const float* __restrict__ dnnout,
                        const float* __restrict__ w,
                        const float* __restrict__ bias,
                        float* __restrict__ out) {
  __shared__ float red[256];
  const int b = blockIdx.x;
  const int t = threadIdx.x;
  float acc = 0.0f;
  for (int i = t; i < CONCAT; i += 256) {
    float v;
    if (i < F_SZ)            v = liner[b * F_SZ + i];
    else if (i < F_SZ + 50)  v = cin[b * 50 + (i - F_SZ)];
    else                     v = dnnout[(size_t)b * (F_SZ * DNNOUT) + (i - F_SZ - 50)];
    acc += v * w[i];
  }
  red[t] = acc;
  __syncthreads();
  for (int s = 128; s > 0; s >>= 1) {
    if (t < s) red[t] += red[t + s];
    __syncthreads();
  }
  if (t == 0) out[b] = red[0] + bias[0];
}

// --------------------------------------------------------------------------
extern "C" void kernel_launch(void* const* d_in, const int* in_sizes, int n_in,
                              void* d_out, int out_size, void* d_ws, size_t ws_size,
                              hipStream_t stream) {
  const float* x    = (const float*)d_in[0];
  const float* embW = (const float*)d_in[1];
  const float* embB = (const float*)d_in[2];
  const float* linW = (const float*)d_in[3];
  const float* linB = (const float*)d_in[4];
  const float* W0   = (const float*)d_in[5];
  const float* b0   = (const float*)d_in[6];
  const float* W1   = (const float*)d_in[7];
  const float* b1   = (const float*)d_in[8];
  const float* Wo   = (const float*)d_in[9];
  const float* bo   = (const float*)d_in[10];
  const float* cw0  = (const float*)d_in[11];
  const float* cw1  = (const float*)d_in[12];
  const float* cw2  = (const float*)d_in[13];
  const float* cw3  = (const float*)d_in[14];
  const float* cw4  = (const float*)d_in[15];
  const float* d1W  = (const float*)d_in[16];
  const float* d1b  = (const float*)d_in[17];

  float* ws     = (float*)d_ws;
  float* embP   = ws;                                  // ROWS*16
  float* liner  = embP + (size_t)ROWS * 16;            // ROWS
  float* dnnout = liner + ROWS;                        // ROWS*32
  float* cin    = dnnout + (size_t)ROWS * DNNOUT;      // B*50
  // total ~26 MB of workspace

  k_emb_liner<<<ROWS / 16 / 8, 256, 0, stream>>>(x, embW, embB, linW, linB, embP, liner);
  k_dnn      <<<ROWS / 16,     128, 0, stream>>>(embP, W0, b0, W1, b1, Wo, bo, dnnout);
  k_cin      <<<B_SZ,          256, 0, stream>>>(embP, cw0, cw1, cw2, cw3, cw4, cin);
  k_final    <<<B_SZ,          256, 0, stream>>>(liner, cin, dnnout, d1W, d1b, (float*)d_out);
}